// GeoPPFDescriptor_66090956751029
// MI455X (gfx1250) — compile-verified
//
#include <hip/hip_runtime.h>
#include <math.h>

#define N_PTS  2048
#define KNN    10
#define OUT_CH 256

typedef __attribute__((ext_vector_type(2))) float v2f;
typedef __attribute__((ext_vector_type(8))) float v8f;
typedef __attribute__((ext_vector_type(4))) int   v4i;

#if __has_builtin(__builtin_amdgcn_global_load_async_to_lds_b128)
#define HAVE_ASYNC_COPY 1
#endif

typedef __attribute__((address_space(1))) char* gchar_p;
typedef __attribute__((address_space(3))) char* lchar_p;
typedef __attribute__((address_space(1))) v4i*  gvec_p;
typedef __attribute__((address_space(3))) v4i*  lvec_p;

__device__ __forceinline__ void wait_async0() {
#if __has_builtin(__builtin_amdgcn_s_wait_asynccnt)
    __builtin_amdgcn_s_wait_asynccnt(0);
#else
    asm volatile("s_wait_asynccnt 0" ::: "memory");
#endif
}

// Stage `nbytes` (multiple of 16*256) from global -> LDS using the CDNA5 async
// memory->LDS path (ASYNCcnt), bypassing VGPRs. Falls back to plain loads.
__device__ __forceinline__ void stage_to_lds(const float* __restrict__ gsrc,
                                             float* lds_dst, int nbytes) {
#ifdef HAVE_ASYNC_COPY
    gchar_p g = (gchar_p)gsrc;      // addrspacecast (drops const; read-only use)
    lchar_p l = (lchar_p)lds_dst;   // addrspacecast generic -> LDS
    for (int off = threadIdx.x * 16; off < nbytes; off += 256 * 16) {
        __builtin_amdgcn_global_load_async_to_lds_b128(
            (gvec_p)(g + off), (lvec_p)(l + off), 0, 0);
    }
    wait_async0();
#else
    const int n = nbytes >> 2;
    for (int idx = threadIdx.x; idx < n; idx += 256) lds_dst[idx] = gsrc[idx];
#endif
}

// ---------------------------------------------------------------------------
// Kernel 1: 10-NN + covariance + smallest eigenvector (per-point normal).
// Whole point cloud staged in LDS (24 KB of the 320 KB WGP LDS).
// Top-11 list kept fully in registers via unrolled bubble-through insertion.
// ---------------------------------------------------------------------------
__global__ __launch_bounds__(256) void knn_normals_kernel(const float* __restrict__ pts,
                                                          float* __restrict__ normals) {
    __shared__ __align__(16) float sp[N_PTS * 3];
    __builtin_prefetch(pts, 0, 1);   // global_prefetch_b8
    stage_to_lds(pts, sp, N_PTS * 3 * 4);
    __syncthreads();

    const int i = blockIdx.x * 256 + threadIdx.x;   // grid sized so i < N_PTS
    const float px = sp[3 * i + 0], py = sp[3 * i + 1], pz = sp[3 * i + 2];

    float bd[KNN + 1];
    int   bi[KNN + 1];
#pragma unroll
    for (int t = 0; t <= KNN; ++t) { bd[t] = 3.0e38f; bi[t] = 0; }

    for (int j = 0; j < N_PTS; ++j) {
        const float dx = px - sp[3 * j + 0];
        const float dy = py - sp[3 * j + 1];
        const float dz = pz - sp[3 * j + 2];
        float dd = dx * dx + dy * dy + dz * dz;
        int jj = j;
        if (dd < bd[KNN]) {
#pragma unroll
            for (int t = 0; t <= KNN; ++t) {
                if (dd < bd[t]) {                     // strict < keeps tie order stable by index
                    const float td = bd[t]; const int ti = bi[t];
                    bd[t] = dd; bi[t] = jj;
                    dd = td;    jj = ti;
                }
            }
        }
    }
    // slot 0 is self (distance 0) -> neighbors are bi[1..KNN]

    const float inv_k = 1.0f / (float)KNN;
    float mx = 0.f, my = 0.f, mz = 0.f;
#pragma unroll
    for (int t = 1; t <= KNN; ++t) {
        const int j = bi[t];
        mx += sp[3 * j + 0]; my += sp[3 * j + 1]; mz += sp[3 * j + 2];
    }
    mx *= inv_k; my *= inv_k; mz *= inv_k;

    float cxx = 0.f, cxy = 0.f, cxz = 0.f, cyy = 0.f, cyz = 0.f, czz = 0.f;
#pragma unroll
    for (int t = 1; t <= KNN; ++t) {
        const int j = bi[t];
        const float ax = sp[3 * j + 0] - mx;
        const float ay = sp[3 * j + 1] - my;
        const float az = sp[3 * j + 2] - mz;
        cxx += ax * ax; cxy += ax * ay; cxz += ax * az;
        cyy += ay * ay; cyz += ay * az; czz += az * az;
    }
    cxx *= inv_k; cxy *= inv_k; cxz *= inv_k; cyy *= inv_k; cyz *= inv_k; czz *= inv_k;

    // Smallest eigenvalue of symmetric 3x3 via trigonometric method.
    const float q  = (cxx + cyy + czz) * (1.0f / 3.0f);
    const float p1 = cxy * cxy + cxz * cxz + cyz * cyz;
    const float b00 = cxx - q, b11 = cyy - q, b22 = czz - q;
    const float p2 = b00 * b00 + b11 * b11 + b22 * b22 + 2.0f * p1;
    float lam;
    if (p2 <= 1e-30f) {
        lam = q;   // ~scalar multiple of identity
    } else {
        const float p    = sqrtf(p2 * (1.0f / 6.0f));
        const float invp = 1.0f / p;
        const float d00 = b00 * invp, d01 = cxy * invp, d02 = cxz * invp;
        const float d11 = b11 * invp, d12 = cyz * invp, d22 = b22 * invp;
        const float detB = d00 * (d11 * d22 - d12 * d12)
                         - d01 * (d01 * d22 - d12 * d02)
                         + d02 * (d01 * d12 - d11 * d02);
        float r = 0.5f * detB;
        r = fminf(1.0f, fmaxf(-1.0f, r));
        const float phi = acosf(r) * (1.0f / 3.0f);
        lam = q + 2.0f * p * cosf(phi + 2.0943951023931953f);   // + 2*pi/3 -> smallest
    }

    // Eigenvector: cross products of rows of (A - lam*I); take best-conditioned pair.
    const float r0x = cxx - lam, r0y = cxy,       r0z = cxz;
    const float r1x = cxy,       r1y = cyy - lam, r1z = cyz;
    const float r2x = cxz,       r2y = cyz,       r2z = czz - lam;
    const float v0x = r0y * r1z - r0z * r1y, v0y = r0z * r1x - r0x * r1z, v0z = r0x * r1y - r0y * r1x;
    const float v1x = r0y * r2z - r0z * r2y, v1y = r0z * r2x - r0x * r2z, v1z = r0x * r2y - r0y * r2x;
    const float v2x = r1y * r2z - r1z * r2y, v2y = r1z * r2x - r1x * r2z, v2z = r1x * r2y - r1y * r2x;
    const float n0 = v0x * v0x + v0y * v0y + v0z * v0z;
    const float n1 = v1x * v1x + v1y * v1y + v1z * v1z;
    const float n2 = v2x * v2x + v2y * v2y + v2z * v2z;
    float vx = v0x, vy = v0y, vz = v0z, nn = n0;
    if (n1 > nn) { vx = v1x; vy = v1y; vz = v1z; nn = n1; }
    if (n2 > nn) { vx = v2x; vy = v2y; vz = v2z; nn = n2; }
    if (nn < 1e-30f) { vx = 1.0f; vy = 0.0f; vz = 0.0f; nn = 1.0f; }
    const float rinv = rsqrtf(nn);
    normals[3 * i + 0] = vx * rinv;
    normals[3 * i + 1] = vy * rinv;
    normals[3 * i + 2] = vz * rinv;
}

// ---------------------------------------------------------------------------
// Kernel 2: PPF feature means. One wave32 per point i; lanes stride over j.
// Points + normals async-staged into LDS (48 KB). Wave shuffle reduction.
// ---------------------------------------------------------------------------
__global__ __launch_bounds__(256) void ppf_feat_kernel(const float* __restrict__ pts,
                                                       const float* __restrict__ normals,
                                                       float* __restrict__ feat) {
    __shared__ __align__(16) float sp[N_PTS * 3];
    __shared__ __align__(16) float sn[N_PTS * 3];
    stage_to_lds(pts,     sp, N_PTS * 3 * 4);
    stage_to_lds(normals, sn, N_PTS * 3 * 4);
    __syncthreads();

    const int lane = threadIdx.x & 31;
    const int wave = threadIdx.x >> 5;
    const int i = blockIdx.x * 8 + wave;

    const float px = sp[3 * i + 0], py = sp[3 * i + 1], pz = sp[3 * i + 2];
    const float ax = sn[3 * i + 0], ay = sn[3 * i + 1], az = sn[3 * i + 2];

    float s0 = 0.f, s1 = 0.f, s2 = 0.f, s3 = 0.f;
    for (int j = lane; j < N_PTS; j += 32) {
        const float dx = px - sp[3 * j + 0];
        const float dy = py - sp[3 * j + 1];
        const float dz = pz - sp[3 * j + 2];
        const float bx = sn[3 * j + 0], by = sn[3 * j + 1], bz = sn[3 * j + 2];

        const float dsq = dx * dx + dy * dy + dz * dz;
        s0 += (dsq > 0.f) ? sqrtf(dsq) : 0.f;
        {   // angle(n_i, d)
            const float cx = ay * dz - az * dy;
            const float cy = az * dx - ax * dz;
            const float cz = ax * dy - ay * dx;
            const float csq = cx * cx + cy * cy + cz * cz;
            const float cn = (csq > 0.f) ? sqrtf(csq) : 0.f;
            s1 += atan2f(cn, ax * dx + ay * dy + az * dz);
        }
        {   // angle(n_j, d)
            const float cx = by * dz - bz * dy;
            const float cy = bz * dx - bx * dz;
            const float cz = bx * dy - by * dx;
            const float csq = cx * cx + cy * cy + cz * cz;
            const float cn = (csq > 0.f) ? sqrtf(csq) : 0.f;
            s2 += atan2f(cn, bx * dx + by * dy + bz * dz);
        }
        {   // angle(n_i, n_j)
            const float cx = ay * bz - az * by;
            const float cy = az * bx - ax * bz;
            const float cz = ax * by - ay * bx;
            const float csq = cx * cx + cy * cy + cz * cz;
            const float cn = (csq > 0.f) ? sqrtf(csq) : 0.f;
            s3 += atan2f(cn, ax * bx + ay * by + az * bz);
        }
    }

#pragma unroll
    for (int off = 16; off > 0; off >>= 1) {
        s0 += __shfl_down(s0, off, 32);
        s1 += __shfl_down(s1, off, 32);
        s2 += __shfl_down(s2, off, 32);
        s3 += __shfl_down(s3, off, 32);
    }
    if (lane == 0) {
        const float invn = 1.0f / (float)N_PTS;
        feat[4 * i + 0] = s0 * invn;
        feat[4 * i + 1] = s1 * invn;
        feat[4 * i + 2] = s2 * invn;
        feat[4 * i + 3] = s3 * invn;
    }
}

// ---------------------------------------------------------------------------
// Kernel 3: out = feat(2048x4) @ W.T(4x256) + b  via V_WMMA_F32_16X16X4_F32.
// One 16x16 output tile per wave32: A = 16x4 feat tile (2 VGPRs/lane),
// B = 4x16 slice of W.T (2 VGPRs/lane), C preloaded with bias.
// ISA §7.12.2 layouts: A lane<16 holds (K0,K1) of row (lane), lane>=16 holds
// (K2,K3); B mirrors over columns; D VGPR v = row v+8*(lane>=16), col lane&15.
// ---------------------------------------------------------------------------
__global__ __launch_bounds__(256) void gemm_wmma_kernel(const float* __restrict__ feat,
                                                        const float* __restrict__ W,
                                                        const float* __restrict__ bias,
                                                        float* __restrict__ out) {
    const int lane  = threadIdx.x & 31;
    const int wave  = threadIdx.x >> 5;
    const int tile  = blockIdx.x * 8 + wave;   // 0..2047  (128 mtiles x 16 ntiles)
    const int mtile = tile >> 4;
    const int ntile = tile & 15;

    const int hl    = lane & 15;
    const int hi    = (lane >> 4) & 1;
    const int kbase = hi * 2;

    const int arow = mtile * 16 + hl;
    v2f a;
    a.x = feat[arow * 4 + kbase + 0];
    a.y = feat[arow * 4 + kbase + 1];

    const int ncol = ntile * 16 + hl;
    v2f bm;
    bm.x = W[ncol * 4 + kbase + 0];            // B[K][N] = W.T[K][N] = W[N][K]
    bm.y = W[ncol * 4 + kbase + 1];

    const float bv = bias[ncol];
    v8f c;
#pragma unroll
    for (int v = 0; v < 8; ++v) c[v] = bv;

    // 8 args: (neg_a, A, neg_b, B, c_mod, C, reuse_a, reuse_b)
    c = __builtin_amdgcn_wmma_f32_16x16x4_f32(false, a, false, bm,
                                              (short)0, c, false, false);

#pragma unroll
    for (int v = 0; v < 8; ++v) {
        const int m = mtile * 16 + hi * 8 + v;
        out[m * OUT_CH + ncol] = c[v];
    }
}

// ---------------------------------------------------------------------------
extern "C" void kernel_launch(void* const* d_in, const int* in_sizes, int n_in,
                              void* d_out, int out_size, void* d_ws, size_t ws_size,
                              hipStream_t stream) {
    const float* pts  = (const float*)d_in[0];   // (1, 2048, 3) f32
    const float* W    = (const float*)d_in[1];   // (256, 4)     f32
    const float* bias = (const float*)d_in[2];   // (256,)       f32
    float* out = (float*)d_out;                  // (1, 2048, 256) f32

    float* normals = (float*)d_ws;               // 2048*3 floats
    float* feat    = normals + N_PTS * 3;        // 2048*4 floats

    knn_normals_kernel<<<N_PTS / 256, 256, 0, stream>>>(pts, normals);
    ppf_feat_kernel<<<N_PTS / 8, 256, 0, stream>>>(pts, normals, feat);
    gemm_wmma_kernel<<<(N_PTS / 16) * (OUT_CH / 16) / 8, 256, 0, stream>>>(feat, W, bias, out);
}